// MultiVarModel_42597485641910
// MI455X (gfx1250) — compile-verified
//
#include <hip/hip_runtime.h>
#include <math.h>

// ---------------- problem constants (from reference) ----------------
#define NPTS     400000
#define JDIM     8
#define LOG2PI_F 1.8378770664093453f
#define LN2_F    0.6931471805599453f

// ---------------- tuning ----------------
#define NBINS    (1 << 18)            // counting-sort bins over t in [0, 3650)
#define SCAN_BLK 1024
#define NBLKS    (NBINS / SCAN_BLK)   // 256
#define NSEG     512                  // parallel scan segments, one wave32 each
#define SEGLEN   ((NPTS + NSEG - 1) / NSEG)
#define WARMUP   4096                 // exponential-forgetting warm-up overlap
#define MCHUNK   4                    // 4 rank-1 updates -> one rank-4 f32 WMMA
#define SLACK    8                    // tail slack rows for unconditional loads

typedef __attribute__((ext_vector_type(2))) float v2f;   // WMMA f32 16x16x4 A/B operand
typedef __attribute__((ext_vector_type(8))) float v8f;   // WMMA f32 C/D accumulator

// ---------------- width-8 lane exchange via DPP8 (VALU-latency, no LDS) ----------
#if __has_builtin(__builtin_amdgcn_mov_dpp8)
template <int SEL>
__device__ __forceinline__ float dpp8_mov(float x) {
  return __int_as_float(__builtin_amdgcn_mov_dpp8(__float_as_int(x), SEL));
}
template <int Q>  // broadcast lane Q within each 8-lane group
__device__ __forceinline__ float bc8(float x) {
  constexpr int sel = (Q) | (Q << 3) | (Q << 6) | (Q << 9) |
                      (Q << 12) | (Q << 15) | (Q << 18) | (Q << 21);
  return dpp8_mov<sel>(x);
}
template <int M>  // lane i reads lane i^M within each 8-lane group
__device__ __forceinline__ float xor8(float x) {
  constexpr int sel = ((0 ^ M)) | ((1 ^ M) << 3) | ((2 ^ M) << 6) | ((3 ^ M) << 9) |
                      ((4 ^ M) << 12) | ((5 ^ M) << 15) | ((6 ^ M) << 18) | ((7 ^ M) << 21);
  return dpp8_mov<sel>(x);
}
#else
template <int Q>
__device__ __forceinline__ float bc8(float x) { return __shfl(x, Q, 8); }
template <int M>
__device__ __forceinline__ float xor8(float x) { return __shfl_xor(x, M, 8); }
#endif

// raw v_log_f32 (log2); accumulate in log2 domain, convert by ln2 once per segment
__device__ __forceinline__ float fast_log2(float x) {
#if __has_builtin(__builtin_amdgcn_logf)
  return __builtin_amdgcn_logf(x);
#else
  return __log2f(x);
#endif
}

__device__ __forceinline__ int t_key(float t) {
  int b = (int)(t * ((float)NBINS / 3650.0f));
  return b < 0 ? 0 : (b >= NBINS ? NBINS - 1 : b);
}

// ---------------------------------------------------------------- sort
__global__ void k_hist(const float* __restrict__ t, unsigned* __restrict__ hist) {
  int i = blockIdx.x * blockDim.x + threadIdx.x;
  if (i >= NPTS) return;
  atomicAdd(&hist[t_key(t[i])], 1u);
}

__global__ void k_scan_blk(unsigned* __restrict__ hist, unsigned* __restrict__ blksum) {
  __shared__ unsigned sh[SCAN_BLK];
  const int tid = threadIdx.x;
  const int g = blockIdx.x * SCAN_BLK + tid;
  unsigned v = hist[g];
  unsigned acc = v;
  sh[tid] = v;
  __syncthreads();
  for (int off = 1; off < SCAN_BLK; off <<= 1) {
    unsigned add = (tid >= off) ? sh[tid - off] : 0u;
    __syncthreads();
    acc += add;
    sh[tid] = acc;
    __syncthreads();
  }
  hist[g] = acc - v;                      // exclusive within block
  if (tid == SCAN_BLK - 1) blksum[blockIdx.x] = acc;
}

__global__ void k_scan_top(unsigned* __restrict__ blksum) {
  __shared__ unsigned sh[NBLKS];
  const int tid = threadIdx.x;
  unsigned v = blksum[tid];
  unsigned acc = v;
  sh[tid] = v;
  __syncthreads();
  for (int off = 1; off < NBLKS; off <<= 1) {
    unsigned add = (tid >= off) ? sh[tid - off] : 0u;
    __syncthreads();
    acc += add;
    sh[tid] = acc;
    __syncthreads();
  }
  blksum[tid] = acc - v;                  // exclusive over block sums
}

__global__ void k_scan_add(unsigned* __restrict__ hist, const unsigned* __restrict__ blksum,
                           unsigned* __restrict__ cursor) {
  int g = blockIdx.x * SCAN_BLK + threadIdx.x;
  unsigned s = hist[g] + blksum[blockIdx.x];
  hist[g] = s;
  cursor[g] = s;                          // running cursor for atomic scatter
}

__global__ void k_scatter(const float* __restrict__ t, const int* __restrict__ band,
                          const float* __restrict__ y, const float* __restrict__ yerr,
                          const float* __restrict__ las, unsigned* __restrict__ cursor,
                          float* __restrict__ t_s, float* __restrict__ y_s,
                          float* __restrict__ diag_s, float* __restrict__ amp_s) {
  int i = blockIdx.x * blockDim.x + threadIdx.x;
  if (i >= NPTS) return;
  float tv = t[i];
  unsigned pos = atomicAdd(&cursor[t_key(tv)], 1u);
  t_s[pos] = tv;
  y_s[pos] = y[i];
  float e = yerr[i];
  diag_s[pos] = e * e;
  int b = band[i];
  amp_s[pos] = (b == 0) ? 1.0f : __expf(las[b - 1]);
}

// ---------------------------------------------------------------- prep
// phi[i][j] = exp(-c_j*dt_i) (row-major, 32B/step); sc4[i] = (amp, A, y, 0)
__global__ void k_prep(const float* __restrict__ lkp, const float* __restrict__ t_s,
                       const float* __restrict__ diag_s, const float* __restrict__ amp_s,
                       const float* __restrict__ y_s,
                       float* __restrict__ phi, float4* __restrict__ sc4) {
  int i = blockIdx.x * blockDim.x + threadIdx.x;
  if (i >= NPTS) return;
  float dt = (i == 0) ? 0.0f : (t_s[i] - t_s[i - 1]);
  float suma = 0.0f;
#pragma unroll
  for (int j = 0; j < JDIM; ++j) {
    suma += __expf(lkp[j]);
    float cj = __expf(lkp[JDIM + j]);
    phi[(size_t)i * JDIM + j] = __expf(-cj * dt);
  }
  float amp = amp_s[i];
  float A = diag_s[i] + amp * amp * suma;
  sc4[i] = make_float4(amp, A, y_s[i], 0.0f);
}

// ---------------------------------------------------------------- scan
// One wave32 per segment. Ghat (q-stabilized S) lives in WMMA f32 C-layout
// (8 VGPRs; VGPR m = row m, lane = col), replicated 2x2 over the 16x16 tile so
// every 8-lane group holds an identical copy: all lane exchange is width-8 DPP8
// and WMMA (EXEC all-ones) preserves the invariant. Every 4 steps the deferred
// rank-1 updates apply as one rank-4 Gram update via v_wmma_f32_16x16x4_f32,
// then Ghat/fhat rescale by the chunk's cumulative phi product (branchless null
// steps pad the tail: phv=1, amp=0, A=1, y=0 contribute exactly zero).
__global__ void k_scanwave(const float* __restrict__ lkp,
                           const float* __restrict__ phi,
                           const float4* __restrict__ sc4,
                           float* __restrict__ partials) {
  const int lane = threadIdx.x & 31;
  int seg = blockIdx.x * (blockDim.x >> 5) + (threadIdx.x >> 5);
  seg = __builtin_amdgcn_readfirstlane(seg);   // force scalar control flow
  if (seg >= NSEG) return;
  const int j = lane & 7;
  const bool hiHalf = lane >= 16;

  const float a_j = __expf(lkp[j]);            // kernel amplitudes a_j

  const int own0 = seg * SEGLEN;
  const int iend = min(NPTS, own0 + SEGLEN);
  const int i0 = max(0, own0 - WARMUP);

  v8f G = {0.f, 0.f, 0.f, 0.f, 0.f, 0.f, 0.f, 0.f};        // Ghat, C-layout
  float f8[8] = {0.f, 0.f, 0.f, 0.f, 0.f, 0.f, 0.f, 0.f};  // fhat, replicated
  float acc_zd = 0.f, acc_l2 = 0.f;            // sum z^2/D, sum log2(D)

  // double-buffered chunk loads (arrays have SLACK rows past NPTS)
  float phvN[MCHUNK];
  float4 scN[MCHUNK];
#pragma unroll
  for (int m = 0; m < MCHUNK; ++m) {
    phvN[m] = phi[(size_t)(i0 + m) * JDIM + j];
    scN[m] = sc4[i0 + m];
  }

  for (int base = i0; base < iend; base += MCHUNK) {
    float phvC[MCHUNK];
    float4 scC[MCHUNK];
#pragma unroll
    for (int m = 0; m < MCHUNK; ++m) { phvC[m] = phvN[m]; scC[m] = scN[m]; }
    // issue next chunk's loads now; they retire while we do this chunk's math
#pragma unroll
    for (int m = 0; m < MCHUNK; ++m) {
      phvN[m] = phi[(size_t)(base + MCHUNK + m) * JDIM + j];
      scN[m] = sc4[base + MCHUNK + m];
    }
    // stay a full cacheline ahead of the pipelined loads
    __builtin_prefetch(phi + (size_t)(base + 4 * MCHUNK) * JDIM, 0, 1);

    float r = 1.0f;                 // cumulative phi product within chunk
    float rinv = 1.0f;              // cumulative reciprocal (v_rcp, no divides)
    float a0 = 0.f, a1 = 0.f;       // A/B-layout panel slots (rows 8-15 replicas)
    float W0[8], W1[8], W2[8];      // prior in-chunk What columns

#pragma unroll
    for (int m = 0; m < MCHUNK; ++m) {
      const int i = base + m;
      const bool live = (i < iend);           // scalar after readfirstlane
      float phv = live ? phvC[m] : 1.0f;      // null step: phi=1
      float amp = live ? scC[m].x : 0.0f;     //            amp=0
      float Ain = live ? scC[m].y : 1.0f;     //            A=1 -> log D = 0
      float yin = live ? scC[m].z : 0.0f;     //            y=0 -> z = 0

      r *= phv;
      rinv *= __builtin_amdgcn_rcpf(phv);
      float uh = r * (amp * a_j);             // uhat_j
      float vh = amp * rinv;                  // vhat_j (V_j == amp)

      // h = Ghat * uhat : 8 parallel width-8 DPP butterflies
      float h[8];
#pragma unroll
      for (int q = 0; q < 8; ++q) h[q] = G[q] * uh;
#pragma unroll
      for (int q = 0; q < 8; ++q) {
        h[q] += xor8<1>(h[q]);
        h[q] += xor8<2>(h[q]);
        h[q] += xor8<4>(h[q]);
      }
      // broadcast uhat / vhat components (DPP8 lane-broadcasts)
      float ub[8], vb[8];
      ub[0] = bc8<0>(uh); vb[0] = bc8<0>(vh);
      ub[1] = bc8<1>(uh); vb[1] = bc8<1>(vh);
      ub[2] = bc8<2>(uh); vb[2] = bc8<2>(vh);
      ub[3] = bc8<3>(uh); vb[3] = bc8<3>(vh);
      ub[4] = bc8<4>(uh); vb[4] = bc8<4>(vh);
      ub[5] = bc8<5>(uh); vb[5] = bc8<5>(vh);
      ub[6] = bc8<6>(uh); vb[6] = bc8<6>(vh);
      ub[7] = bc8<7>(uh); vb[7] = bc8<7>(vh);

      // right-looking corrections for rank-1s deferred inside this chunk
      if (m >= 1) {
        float s = 0.f;
#pragma unroll
        for (int q = 0; q < 8; ++q) s += W0[q] * ub[q];
#pragma unroll
        for (int q = 0; q < 8; ++q) h[q] += s * W0[q];
      }
      if (m >= 2) {
        float s = 0.f;
#pragma unroll
        for (int q = 0; q < 8; ++q) s += W1[q] * ub[q];
#pragma unroll
        for (int q = 0; q < 8; ++q) h[q] += s * W1[q];
      }
      if (m >= 3) {
        float s = 0.f;
#pragma unroll
        for (int q = 0; q < 8; ++q) s += W2[q] * ub[q];
#pragma unroll
        for (int q = 0; q < 8; ++q) h[q] += s * W2[q];
      }

      float dD = 0.f, zq = 0.f;
#pragma unroll
      for (int q = 0; q < 8; ++q) { dD += ub[q] * h[q]; zq += ub[q] * f8[q]; }
      float D = fmaxf(Ain - dD, 1e-18f);
      float z = yin - zq;
      float rs = __builtin_amdgcn_rsqf(D);    // v_rsq_f32

      float Wv[8];                            // What column (full vector/lane)
#pragma unroll
      for (int q = 0; q < 8; ++q) Wv[q] = (vb[q] - h[q]) * rs;
      float zs = z * rs;
#pragma unroll
      for (int q = 0; q < 8; ++q) f8[q] += Wv[q] * zs;

      float w = (i >= own0) ? 1.0f : 0.0f;    // discard warm-up outputs
      acc_zd += w * zs * zs;                  // z^2 / D
      acc_l2 += w * fast_log2(D);             // raw v_log_f32; *ln2 at the end

      // own component What[l&7] via static select tree
      float t0 = (j & 1) ? h[1] : h[0];
      float t1 = (j & 1) ? h[3] : h[2];
      float t2 = (j & 1) ? h[5] : h[4];
      float t3 = (j & 1) ? h[7] : h[6];
      float u0 = (j & 2) ? t1 : t0;
      float u1 = (j & 2) ? t3 : t2;
      float h_own = (j & 4) ? u1 : u0;
      float Wown = (vh - h_own) * rs;

      // stash into panel (lanes<16: K=0,1 ; lanes>=16: K=2,3)
      if (m == 0) {
#pragma unroll
        for (int q = 0; q < 8; ++q) W0[q] = Wv[q];
        if (!hiHalf) a0 = Wown;
      } else if (m == 1) {
#pragma unroll
        for (int q = 0; q < 8; ++q) W1[q] = Wv[q];
        if (!hiHalf) a1 = Wown;
      } else if (m == 2) {
#pragma unroll
        for (int q = 0; q < 8; ++q) W2[q] = Wv[q];
        if (hiHalf) a0 = Wown;
      } else {
        if (hiHalf) a1 = Wown;
      }
    }

    // rank-4 Gram update on the matrix pipe: Ghat += Wpan * Wpan^T
    v2f Apan = {a0, a1};
#if __has_builtin(__builtin_amdgcn_wmma_f32_16x16x4_f32)
    G = __builtin_amdgcn_wmma_f32_16x16x4_f32(false, Apan, false, Apan,
                                              (short)0, G, false, false);
#else
    {
      float s0 = Apan.x, s1 = Apan.y;
#pragma unroll
      for (int q = 0; q < 8; ++q) {
        float c0 = __shfl(s0, q, 8), c1 = __shfl(s1, q, 8);
        G[q] += s0 * c0 + s1 * c1;
      }
    }
#endif

    // Ghat <- diag(r) Ghat diag(r);  fhat <- r .* fhat
    float rho = r;
    float rb0 = bc8<0>(rho), rb1 = bc8<1>(rho), rb2 = bc8<2>(rho), rb3 = bc8<3>(rho);
    float rb4 = bc8<4>(rho), rb5 = bc8<5>(rho), rb6 = bc8<6>(rho), rb7 = bc8<7>(rho);
    G[0] *= rb0 * rho; f8[0] *= rb0;
    G[1] *= rb1 * rho; f8[1] *= rb1;
    G[2] *= rb2 * rho; f8[2] *= rb2;
    G[3] *= rb3 * rho; f8[3] *= rb3;
    G[4] *= rb4 * rho; f8[4] *= rb4;
    G[5] *= rb5 * rho; f8[5] *= rb5;
    G[6] *= rb6 * rho; f8[6] *= rb6;
    G[7] *= rb7 * rho; f8[7] *= rb7;
  }

  if (lane == 0) {
    partials[seg] = acc_zd;
    partials[NSEG + seg] = acc_l2 * LN2_F;     // log2 -> ln once per segment
  }
}

// ---------------------------------------------------------------- final reduce
__global__ void k_final(const float* __restrict__ partials, float* __restrict__ out) {
  __shared__ float sh[256];
  float s = 0.f;
  for (int i = threadIdx.x; i < 2 * NSEG; i += 256) s += partials[i];
  sh[threadIdx.x] = s;
  __syncthreads();
  for (int off = 128; off > 0; off >>= 1) {
    if (threadIdx.x < off) sh[threadIdx.x] += sh[threadIdx.x + off];
    __syncthreads();
  }
  if (threadIdx.x == 0)
    out[0] = -0.5f * (sh[0] + (float)NPTS * LOG2PI_F);
}

// ---------------------------------------------------------------- launch
extern "C" void kernel_launch(void* const* d_in, const int* in_sizes, int n_in,
                              void* d_out, int out_size, void* d_ws, size_t ws_size,
                              hipStream_t stream) {
  (void)in_sizes; (void)n_in; (void)out_size; (void)ws_size;
  const float* t    = (const float*)d_in[0];
  const int*   band = (const int*)d_in[1];
  const float* y    = (const float*)d_in[2];
  const float* yerr = (const float*)d_in[3];
  const float* lkp  = (const float*)d_in[4];
  const float* las  = (const float*)d_in[5];
  float* out = (float*)d_out;

  // workspace carve-up (~28 MB total)
  char* ws = (char*)d_ws;
  size_t off = 0;
  auto carve = [&](size_t bytes) -> void* {
    void* p = (void*)(ws + off);
    off += (bytes + 255) & ~(size_t)255;
    return p;
  };
  unsigned* hist   = (unsigned*)carve(sizeof(unsigned) * NBINS);
  unsigned* cursor = (unsigned*)carve(sizeof(unsigned) * NBINS);
  unsigned* blksum = (unsigned*)carve(sizeof(unsigned) * NBLKS);
  float* t_s    = (float*)carve(sizeof(float) * NPTS);
  float* y_s    = (float*)carve(sizeof(float) * NPTS);
  float* diag_s = (float*)carve(sizeof(float) * NPTS);
  float* amp_s  = (float*)carve(sizeof(float) * NPTS);
  float* phiarr = (float*)carve(sizeof(float) * (NPTS + SLACK) * JDIM);
  float4* sc4   = (float4*)carve(sizeof(float4) * (NPTS + SLACK));
  float* parts  = (float*)carve(sizeof(float) * 2 * NSEG);

  const int TPB = 256;
  const int NB = (NPTS + TPB - 1) / TPB;

  hipMemsetAsync(hist, 0, sizeof(unsigned) * NBINS, stream);
  k_hist<<<NB, TPB, 0, stream>>>(t, hist);
  k_scan_blk<<<NBLKS, SCAN_BLK, 0, stream>>>(hist, blksum);
  k_scan_top<<<1, NBLKS, 0, stream>>>(blksum);
  k_scan_add<<<NBLKS, SCAN_BLK, 0, stream>>>(hist, blksum, cursor);
  k_scatter<<<NB, TPB, 0, stream>>>(t, band, y, yerr, las, cursor,
                                    t_s, y_s, diag_s, amp_s);
  k_prep<<<NB, TPB, 0, stream>>>(lkp, t_s, diag_s, amp_s, y_s, phiarr, sc4);
  k_scanwave<<<NSEG / 8, 256, 0, stream>>>(lkp, phiarr, sc4, parts);
  k_final<<<1, 256, 0, stream>>>(parts, out);
}